// EquivariantDense_55370718380273
// MI455X (gfx1250) — compile-verified
//
#include <hip/hip_runtime.h>
#include <cstdint>

typedef __attribute__((ext_vector_type(2))) float v2f;
typedef __attribute__((ext_vector_type(4))) float v4f;
typedef __attribute__((ext_vector_type(8))) float v8f;

#define B_     8
#define O4_    1024
#define K_     4096
#define N_     16384
#define KCHUNK 512            // k elements per chunk
#define NKC    8              // K_/KCHUNK
#define ROWT   16             // rows per WMMA tile
#define NSLOT  32             // j(4) * kc(8)
#define OUTEL  (B_ * 4 * O4_) // 32768

// Main kernel: one wave handles (b, 16-row tile, j, k-chunk).
// D(16x16,f32) += A(16x4 weights) * B(4x16 x-segments) via v_wmma_f32_16x16x4_f32.
__global__ __launch_bounds__(256) void ed_main(const float* __restrict__ x,
                                               const float* __restrict__ w1,
                                               const float* __restrict__ w2,
                                               const float* __restrict__ w3,
                                               const float* __restrict__ w4,
                                               float* __restrict__ ws) {
  __shared__ float xs[16 * KCHUNK]; // 32 KB: segments 0..3 real, 4..15 zero

  const int tid  = threadIdx.x;
  const int gid  = blockIdx.x;        // 0..2047
  const int b    = gid >> 8;          // 0..7
  const int rem  = gid & 255;
  const int kc   = rem >> 5;          // 0..7  k-chunk
  const int tg   = rem & 31;          // 0..31 tile-group (2 row tiles)
  const int wave = tid >> 5;          // wave32: 8 waves per WG
  const int ln   = tid & 31;
  const int j    = wave & 3;
  const int rt   = tg * 2 + (wave >> 2); // row tile 0..63
  const int obase = rt * ROWT;
  const int kbase = kc * KCHUNK;

  // ---- Stage x chunk into LDS (async global->LDS) + zero padding ----
  // zero-fill segments 4..15 (WMMA columns 4..15 contribute exact zeros)
  for (int idx = tid; idx < 12 * KCHUNK; idx += 256)
    xs[4 * KCHUNK + idx] = 0.0f;
  {
    const float* xb = x + (size_t)b * N_ + kbase;
    #pragma unroll
    for (int q = 0; q < 2; ++q) {
      int f4   = tid + q * 256;   // float4 index 0..511
      int fidx = f4 * 4;          // element index into xs
      int seg    = fidx >> 9;     // /512
      int within = fidx & 511;
      const float* src = xb + (size_t)seg * K_ + within;
      uint32_t ldsoff = (uint32_t)(uintptr_t)(&xs[fidx]);
      asm volatile("global_load_async_to_lds_b128 %0, %1, off"
                   :: "v"(ldsoff), "v"(src) : "memory");
    }
    asm volatile("s_wait_asynccnt 0" ::: "memory");
  }
  __syncthreads();

  // ---- Stream weights (non-temporal: 537 MB single-use vs 192 MB L2),
  //      accumulate with f32 WMMA ----
  const float* W = (j == 0) ? w1 : (j == 1) ? w2 : (j == 2) ? w3 : w4;
  const int r  = ln & 15;           // A row (and B column) for this lane
  const int kk = (ln >> 4) * 2;     // k sub-offset: lanes 0-15 -> k0,k1; 16-31 -> k2,k3
  const float* ap = W + ((size_t)b * O4_ + (size_t)(obase + r)) * (size_t)K_
                      + (size_t)(kbase + kk);
  const float* bp = &xs[r * KCHUNK + kk];

  v8f acc = {0.f, 0.f, 0.f, 0.f, 0.f, 0.f, 0.f, 0.f};
  #pragma unroll 8
  for (int t = 0; t < KCHUNK / 4; ++t) {
    v2f a  = __builtin_nontemporal_load((const v2f*)(ap + 4 * t)); // A (global, NT)
    v2f bv = *(const v2f*)(bp + 4 * t);                            // B (LDS)
    acc = __builtin_amdgcn_wmma_f32_16x16x4_f32(
        /*neg_a=*/false, a, /*neg_b=*/false, bv,
        /*c_mod=*/(short)0, acc, /*reuse_a=*/false, /*reuse_b=*/false);
  }

  // ---- Write useful columns (s = n < 4) to workspace slot (j, kc);
  //      default temporal policy so ws stays L2-resident for ed_reduce ----
  const int n = ln & 15;
  if (n < 4) {
    const int i     = (n - j) & 3;      // output block index
    const int mbase = (ln >> 4) * 8;    // rows 0..7 (lanes<16) or 8..15
    const int slot  = j * NKC + kc;
    float* dst = ws + (size_t)slot * OUTEL + (size_t)b * (4 * O4_)
               + (size_t)i * O4_ + obase + mbase;
    v4f lo = {acc[0], acc[1], acc[2], acc[3]};
    v4f hi = {acc[4], acc[5], acc[6], acc[7]};
    *(v4f*)(dst)     = lo;
    *(v4f*)(dst + 4) = hi;
  }
}

// Deterministic fixed-order reduction over the 32 (j, kc) slots (ws is L2-hot).
__global__ __launch_bounds__(256) void ed_reduce(const float* __restrict__ ws,
                                                 float* __restrict__ out) {
  int e = blockIdx.x * 256 + threadIdx.x; // 0..32767
  float s = 0.f;
  #pragma unroll
  for (int slot = 0; slot < NSLOT; ++slot)
    s += ws[(size_t)slot * OUTEL + e];
  out[e] = s;
}

extern "C" void kernel_launch(void* const* d_in, const int* in_sizes, int n_in,
                              void* d_out, int out_size, void* d_ws, size_t ws_size,
                              hipStream_t stream) {
  (void)in_sizes; (void)n_in; (void)out_size; (void)ws_size;
  const float* x  = (const float*)d_in[0];
  const float* w1 = (const float*)d_in[1];
  const float* w2 = (const float*)d_in[2];
  const float* w3 = (const float*)d_in[3];
  const float* w4 = (const float*)d_in[4];
  float* ws  = (float*)d_ws;   // needs 32 * 32768 * 4 B = 4 MiB
  float* out = (float*)d_out;

  ed_main<<<dim3(2048), dim3(256), 0, stream>>>(x, w1, w2, w3, w4, ws);
  ed_reduce<<<dim3(OUTEL / 256), dim3(256), 0, stream>>>(ws, out);
}